// UpFIRDn2d_28587302322531
// MI455X (gfx1250) — compile-verified
//
#include <hip/hip_runtime.h>
#include <hip/hip_bf16.h>

typedef float v2f __attribute__((ext_vector_type(2)));
typedef float v8f __attribute__((ext_vector_type(8)));

#define B_    16
#define C_    64
#define RES   128
#define BANDS 8          // 8 bands of 16 output rows

// ---- LDS layout (float offsets), phase-aliased. Total 13232 floats = 52928 B ----
// Segment S0 (always live): vc 48x137
#define OFF_VC   0
#define VC_W     137                // odd pitch: conflict-free WMMA-A reads
// Segment S1: phase A: xs 32x128 | phase B: y_blk 16x256 | phase C: w 16x273 + T 44x16
#define OFF_XS   6576               // 32x128 = 4096
#define OFF_YBLK 6576               // 16x256 = 4096
#define OFF_W    6576               // 16x273 = 4368
#define W_W      273
#define OFF_T    10944              // 44x16 = 704      (S1 ends at 11648)
// Segment S2: phase A: P 48x33 | phase B: U 16x16 + Q 16x49
#define OFF_P    11648              // 48x33 = 1584
#define P_W      33
#define OFF_U    11648              // 16x16 = 256
#define OFF_Q    11904              // 16x49 = 784      (S2 ends at 13232)
#define Q_W      49
#define SMEM_F   13232

#define WMMA_F32X4 __builtin_amdgcn_wmma_f32_16x16x4_f32

__global__ __launch_bounds__(256)
void upfirdn2d_fused_kernel(const float* __restrict__ x,
                            const float* __restrict__ bias,
                            const float* __restrict__ upf,
                            const float* __restrict__ dnf,
                            float* __restrict__ out)
{
    __shared__ float sm[SMEM_F];
    const int tid  = threadIdx.x;
    const int lane = tid & 31;
    const int wave = tid >> 5;

    const int blk  = blockIdx.x;
    const int bc   = blk >> 3;          // channel-image 0..1023
    const int band = blk & 7;
    const int m0   = band * 16;
    const float bv = bias[bc & (C_ - 1)];

    // ---- recover 1-D taps from rank-1 2-D filters (rcp-mul: no divide storm) ----
    float hu[12], g[13];
    {
        const float rhc = 1.0f / sqrtf(upf[5 * 12 + 5]);
        #pragma unroll
        for (int i = 0; i < 12; ++i) hu[i] = upf[i * 12 + 5] * rhc;
        const float rdc = 1.0f / sqrtf(dnf[5 * 12 + 5]);
        float hd[12];
        #pragma unroll
        for (int i = 0; i < 12; ++i) hd[i] = dnf[i * 12 + 5] * rdc;
        g[0] = 0.5f * hd[0];
        #pragma unroll
        for (int t = 1; t < 12; ++t) g[t] = 0.5f * (hd[t] + hd[t - 1]);
        g[12] = 0.5f * hd[11];
    }

    const int mrow  = lane & 15;          // M (and N for B-frags) within 16x16 tile
    const int khalf = (lane >> 4) << 1;   // lanes 16-31 hold K+2 (f32 16x4 layout)
    const int rhi8  = (lane >> 4) << 3;   // D: VGPR v -> M = v + 8*(lane>=16)

    // ================= phase A: build P, zero vc halos, load x band =============
    // P[rl][s] = sum_i hu[i] * [ (rl+i)>>1 == s ]  (vertical polyphase Toeplitz)
    if (tid < 48) {
        #pragma unroll
        for (int s = 0; s < P_W; ++s) sm[OFF_P + tid * P_W + s] = 0.0f;
        #pragma unroll
        for (int i = 0; i < 12; ++i)
            sm[OFF_P + tid * P_W + ((tid + i) >> 1)] += hu[i];
    }
    // vc column halos (cols 0..2 and 131..136) = 0
    for (int e = tid; e < 48 * 9; e += 256) {
        const int r = e / 9, c9 = e - r * 9;
        sm[OFF_VC + r * VC_W + (c9 < 3 ? c9 : 128 + c9)] = 0.0f;
    }
    // xs rows m0-5 .. m0+24 (+bias), zero-padded; rows 30,31 zeroed (K=32 pad).
    // float4-wide: 4x global_load_b128 + 4x ds_store_b128 per thread.
    {
        const int xlo = m0 - 5;
        const float4* xp4 = (const float4*)(x + (size_t)bc * (RES * RES));
        for (int e = tid; e < 32 * 32; e += 256) {
            const int j = e >> 5, q4 = e & 31;
            const int xr = xlo + j;
            float4 v = make_float4(0.0f, 0.0f, 0.0f, 0.0f);
            if (j < 30 && xr >= 0 && xr < RES) {
                v = xp4[xr * 32 + q4];
                v.x += bv; v.y += bv; v.z += bv; v.w += bv;
            }
            *(float4*)&sm[OFF_XS + j * 128 + q4 * 4] = v;
        }
    }
    __syncthreads();

    // ============ stage 2: vc = P x xs  (M=48, N=128, K=32) — WMMA ==============
    #pragma unroll
    for (int j = 0; j < 3; ++j) {                  // wave owns col-tile nt=wave, 3 row-tiles
        const int mt = j, nt = wave;
        v8f acc = {0.f, 0.f, 0.f, 0.f, 0.f, 0.f, 0.f, 0.f};
        #pragma unroll
        for (int ks = 0; ks < 8; ++ks) {
            const int k0 = ks * 4 + khalf;
            v2f a, b;
            a.x = sm[OFF_P + (mt * 16 + mrow) * P_W + k0];
            a.y = sm[OFF_P + (mt * 16 + mrow) * P_W + k0 + 1];
            b.x = sm[OFF_XS + k0 * 128 + nt * 16 + mrow];
            b.y = sm[OFF_XS + (k0 + 1) * 128 + nt * 16 + mrow];
            acc = WMMA_F32X4(false, a, false, b, (short)0, acc, false, false);
        }
        #pragma unroll
        for (int v = 0; v < 8; ++v)
            sm[OFF_VC + (mt * 16 + v + rhi8) * VC_W + 3 + nt * 16 + mrow] = acc[v];
    }
    __syncthreads();

    // ============ build U (up horizontal Toeplitz) and Q (dn vertical) ==========
    if (tid < 16) {
        // U[k][c] = sum_j hu[j] * [ ((c+j+1)>>1) == k ]  (column-exclusive)
        #pragma unroll
        for (int k = 0; k < 16; ++k) sm[OFF_U + k * 16 + tid] = 0.0f;
        #pragma unroll
        for (int jj = 0; jj < 12; ++jj)
            sm[OFF_U + (((tid + jj + 1) >> 1)) * 16 + tid] += hu[jj];
        // Q[m][r] = g[r - 2m] (row-exclusive)
        #pragma unroll
        for (int r = 0; r < 48; ++r) sm[OFF_Q + tid * Q_W + r] = 0.0f;
        #pragma unroll
        for (int u = 0; u < 13; ++u)
            sm[OFF_Q + tid * Q_W + 2 * tid + u] = g[u];
    }
    __syncthreads();

    // ====== 3 y-blocks: y = lrelu(vc x U)*gain, then w_acc += Q_blk x y =========
    v8f wacc[2];
    #pragma unroll
    for (int it = 0; it < 2; ++it)
        wacc[it] = (v8f){0.f, 0.f, 0.f, 0.f, 0.f, 0.f, 0.f, 0.f};

    for (int blkr = 0; blkr < 3; ++blkr) {
        const int rb = blkr * 16;
        #pragma unroll
        for (int it = 0; it < 2; ++it) {
            const int c0 = (wave * 2 + it) * 16;
            v8f acc = {0.f, 0.f, 0.f, 0.f, 0.f, 0.f, 0.f, 0.f};
            #pragma unroll
            for (int ks = 0; ks < 4; ++ks) {
                const int k0 = ks * 4 + khalf;
                v2f a, b;
                a.x = sm[OFF_VC + (rb + mrow) * VC_W + (c0 >> 1) + k0];
                a.y = sm[OFF_VC + (rb + mrow) * VC_W + (c0 >> 1) + k0 + 1];
                b.x = sm[OFF_U + k0 * 16 + mrow];
                b.y = sm[OFF_U + (k0 + 1) * 16 + mrow];
                acc = WMMA_F32X4(false, a, false, b, (short)0, acc, false, false);
            }
            #pragma unroll
            for (int v = 0; v < 8; ++v) {            // leaky_relu * sqrt(2)
                const float z = acc[v];
                sm[OFF_YBLK + (v + rhi8) * 256 + c0 + mrow] =
                    (z >= 0.0f ? z : 0.2f * z) * 1.41421356237f;
            }
        }
        __syncthreads();
        // w_acc (registers) += Q[:, rb:rb+16] x y_blk
        #pragma unroll
        for (int it = 0; it < 2; ++it) {
            const int c0 = (wave * 2 + it) * 16;
            v8f acc = wacc[it];
            #pragma unroll
            for (int ks = 0; ks < 4; ++ks) {
                const int k0 = ks * 4 + khalf;
                v2f a, b;
                a.x = sm[OFF_Q + mrow * Q_W + rb + k0];
                a.y = sm[OFF_Q + mrow * Q_W + rb + k0 + 1];
                b.x = sm[OFF_YBLK + k0 * 256 + c0 + mrow];
                b.y = sm[OFF_YBLK + (k0 + 1) * 256 + c0 + mrow];
                acc = WMMA_F32X4(false, a, false, b, (short)0, acc, false, false);
            }
            wacc[it] = acc;
        }
        __syncthreads();                              // before y_blk is overwritten
    }

    // ====== phase C: spill w to LDS (+zero pads), build T, dn WMMA ==============
    #pragma unroll
    for (int it = 0; it < 2; ++it) {
        const int c0 = (wave * 2 + it) * 16;
        #pragma unroll
        for (int v = 0; v < 8; ++v)
            sm[OFF_W + (v + rhi8) * W_W + 5 + c0 + mrow] = wacc[it][v];
    }
    for (int e = tid; e < 16 * 17; e += 256) {        // w pads: cols 0..4, 261..272
        const int r = e / 17, c = e - r * 17;
        sm[OFF_W + r * W_W + (c < 5 ? c : 256 + c)] = 0.0f;
    }
    if (tid < 16) {                                    // T[kc][n] = g[kc-2n]
        #pragma unroll
        for (int kc = 0; kc < 44; ++kc) sm[OFF_T + kc * 16 + tid] = 0.0f;
        #pragma unroll
        for (int u = 0; u < 13; ++u)
            sm[OFF_T + (2 * tid + u) * 16 + tid] = g[u];
    }
    __syncthreads();

    {
        const int n0 = wave * 16;
        v8f acc = {0.f, 0.f, 0.f, 0.f, 0.f, 0.f, 0.f, 0.f};
        #pragma unroll
        for (int ks = 0; ks < 11; ++ks) {              // K = 44
            const int k0 = ks * 4 + khalf;
            v2f a, b;
            a.x = sm[OFF_W + mrow * W_W + 2 * n0 + k0];
            a.y = sm[OFF_W + mrow * W_W + 2 * n0 + k0 + 1];
            b.x = sm[OFF_T + k0 * 16 + mrow];
            b.y = sm[OFF_T + (k0 + 1) * 16 + mrow];
            acc = WMMA_F32X4(false, a, false, b, (short)0, acc, false, false);
        }
        float* op = out + (size_t)bc * (RES * RES);
        #pragma unroll
        for (int v = 0; v < 8; ++v)
            op[(m0 + v + rhi8) * RES + n0 + mrow] = acc[v];
    }
}

extern "C" void kernel_launch(void* const* d_in, const int* in_sizes, int n_in,
                              void* d_out, int out_size, void* d_ws, size_t ws_size,
                              hipStream_t stream) {
    (void)in_sizes; (void)n_in; (void)d_ws; (void)ws_size; (void)out_size;
    const float* x    = (const float*)d_in[0];
    const float* bias = (const float*)d_in[1];
    const float* upf  = (const float*)d_in[2];
    const float* dnf  = (const float*)d_in[3];
    float* out = (float*)d_out;

    dim3 grid(B_ * C_ * BANDS), block(256);
    hipLaunchKernelGGL(upfirdn2d_fused_kernel, grid, block, 0, stream,
                       x, bias, upf, dnf, out);
}